// IoU_14250701488677
// MI455X (gfx1250) — compile-verified
//
#include <hip/hip_runtime.h>
#include <hip/hip_bf16.h>

// ---------------------------------------------------------------------------
// YOLO NMS-matching pipeline for MI455X (gfx1250, wave32, WMMA).
//   score (HBM stream, vectorized + global_prefetch_b8)
//   top-K via radix-select histogram + single bitonic sort
//   NMS: parallel suppression-bit matrix + single-wave pipelined sweep
//   cross-match: v_wmma_f32_16x16x4_f32 outer-sum denominators co-executing
//                with the VALU min/max intersection chain, v_rcp_f32 IoU,
//                fully branchless inner loop (cndmask-only masking)
// ---------------------------------------------------------------------------

typedef __attribute__((ext_vector_type(2))) float v2f;
typedef __attribute__((ext_vector_type(4))) float v4f;
typedef __attribute__((ext_vector_type(8))) float v8f;
typedef float f4a4 __attribute__((ext_vector_type(4), aligned(4)));

#define NB 8
#define NN 25200
#define NCLS 80
#define ROW 85
#define KP 1024
#define KC 256
#define IOU_THR 0.45f
#define MAX_WH 4096.0f
#define INV_GN (1.0f / 640.0f)
#define NBINS 2048

// ---------------------------------------------------------------------------
// Stage 1: streaming score. conf = obj * max(cls); mscore = valid ? conf : -1.
// 137 MB of reads -> HBM-bound. b128-capable loads (DWORD-aligned rows are
// legal in DWORD alignment mode) + speculative prefetch ahead.
// ---------------------------------------------------------------------------
__global__ void score_kernel(const float* __restrict__ pred,
                             float* __restrict__ mscore,
                             float conf_thres, int total) {
  int i = blockIdx.x * blockDim.x + threadIdx.x;
  if (i >= total) return;
  const float* r = pred + (size_t)i * ROW;
  __builtin_prefetch(r + ROW * 64, 0, 1);  // global_prefetch_b8, OOB-safe
  f4a4 v0 = *(const f4a4*)(r + 4);         // obj, cls[0..2]
  float obj = v0.x;
  float m = fmaxf(fmaxf(v0.y, v0.z), v0.w);
#pragma unroll
  for (int c = 8; c <= 80; c += 4) {
    f4a4 u = *(const f4a4*)(r + c);
    m = fmaxf(m, fmaxf(fmaxf(u.x, u.y), fmaxf(u.z, u.w)));
  }
  m = fmaxf(m, r[84]);
  float conf = m * obj;
  mscore[i] = ((obj > conf_thres) && (conf > conf_thres)) ? conf : -1.0f;
}

// ---------------------------------------------------------------------------
// Bitonic sort of 1024 (value,index) pairs in LDS. 1024 threads.
// ---------------------------------------------------------------------------
__device__ __forceinline__ void bsort1024(float* v, int* idx, int t,
                                          bool descending) {
  for (int k = 2; k <= 1024; k <<= 1) {
    for (int j = k >> 1; j > 0; j >>= 1) {
      __syncthreads();
      int p = t ^ j;
      if (p > t) {
        bool dirAsc = (((t & k) == 0) != descending);
        float a = v[t], b = v[p];
        bool sw = dirAsc ? (a > b) : (a < b);
        if (sw) {
          v[t] = b; v[p] = a;
          int ia = idx[t]; idx[t] = idx[p]; idx[p] = ia;
        }
      }
    }
  }
  __syncthreads();
}

// ---------------------------------------------------------------------------
// Stage 2: per-image top-1024 via radix select (2048-bin histogram, reversed
// prefix scan for the threshold bin, deterministic scan-ranked tie selection)
// then ONE bitonic sort. Clean path consumes the first 256 entries.
// ---------------------------------------------------------------------------
__global__ void topk_kernel(const float* __restrict__ mscore,
                            int* __restrict__ idx_out) {
  __shared__ int hist[NBINS];
  __shared__ int sc0[NBINS];
  __shared__ int sc1[NBINS];
  __shared__ float sv[KP];
  __shared__ int si[KP];
  __shared__ int cHigh;
  __shared__ int sT, sBase;

  int b = blockIdx.x, t = threadIdx.x;  // 1024 threads
  const float* s = mscore + (size_t)b * NN;
  hist[t] = 0; hist[t + 1024] = 0;
  __syncthreads();

  float myval[25]; int mybin[25];
#pragma unroll
  for (int k = 0; k < 25; ++k) {
    int i = t + (k << 10);
    float v = (i < NN) ? s[i] : -1.0f;
    int bin = (v < 0.0f) ? 0 : min(NBINS - 1, 1 + (int)(v * 2046.0f));
    myval[k] = v; mybin[k] = bin;
    if (i < NN) atomicAdd(&hist[bin], 1);
  }
  __syncthreads();

  // reversed inclusive scan: sc[r] = #elements with bin >= (2047 - r)
  sc0[t] = hist[2047 - t];
  sc0[t + 1024] = hist[1023 - t];
  __syncthreads();
  int* src = sc0; int* dst = sc1;
  for (int d = 1; d < NBINS; d <<= 1) {
    int i0 = t, i1 = t + 1024;
    int a0 = src[i0] + ((i0 >= d) ? src[i0 - d] : 0);
    int a1 = src[i1] + ((i1 >= d) ? src[i1 - d] : 0);
    dst[i0] = a0; dst[i1] = a1;
    __syncthreads();
    int* tmp = src; src = dst; dst = tmp;
  }
  {  // find smallest r with scan[r] >= KP  ->  threshold bin T = 2047 - r
    int i0 = t, i1 = t + 1024;
    int a0 = src[i0], p0 = (i0 > 0) ? src[i0 - 1] : 0;
    if (a0 >= KP && p0 < KP) { sT = 2047 - i0; sBase = p0; }
    int a1 = src[i1], p1 = src[i1 - 1];
    if (a1 >= KP && p1 < KP) { sT = 2047 - i1; sBase = p1; }
    if (t == 0) cHigh = 0;
  }
  __syncthreads();
  int T = sT, basePos = sBase;

  // deterministic rank for threshold-bin ties: per-thread count + block scan
  int myTies = 0;
#pragma unroll
  for (int k = 0; k < 25; ++k)
    myTies += ((mybin[k] == T) && ((t + (k << 10)) < NN)) ? 1 : 0;
  __syncthreads();
  sc0[t] = myTies;
  __syncthreads();
  int* a = sc0; int* d2 = sc1;
  for (int dd = 1; dd < 1024; dd <<= 1) {
    int v = a[t] + ((t >= dd) ? a[t - dd] : 0);
    d2[t] = v;
    __syncthreads();
    int* tmp = a; a = d2; d2 = tmp;
  }
  int tieOff = a[t] - myTies;  // exclusive prefix
  __syncthreads();

  // compaction: all >T taken (order fixed by later sort), ==T rank-limited
  int local = 0;
#pragma unroll
  for (int k = 0; k < 25; ++k) {
    int i = t + (k << 10);
    if (i < NN) {
      int bin = mybin[k];
      if (bin > T) {
        int p = atomicAdd(&cHigh, 1);
        sv[p] = myval[k]; si[p] = i;
      } else if (bin == T) {
        int p = basePos + tieOff + local; local++;
        if (p < KP) { sv[p] = myval[k]; si[p] = i; }
      }
    }
  }
  __syncthreads();
  bsort1024(sv, si, t, /*descending=*/true);
  idx_out[b * KP + t] = si[t];
}

// ---------------------------------------------------------------------------
// Stage 3: gather top-K rows, recompute cls/valid, xywh -> xyxy; also emit
// class-offset boxes + areas for the NMS pair kernel. blockDim = K.
// ---------------------------------------------------------------------------
__global__ void gather_kernel(const float* __restrict__ pred,
                              const int* __restrict__ idx,
                              float conf_thres,
                              float* __restrict__ boxes,
                              float* __restrict__ cls,
                              int* __restrict__ valid,
                              float* __restrict__ obox,
                              float* __restrict__ oarea) {
  int b = blockIdx.x, t = threadIdx.x;
  int K = blockDim.x;
  int i = idx[b * KP + t];
  const float* r = pred + ((size_t)b * NN + i) * ROW;
  float obj = r[4];
  float m = -1e30f; int mi = 0;
#pragma unroll 4
  for (int c = 0; c < NCLS; ++c) {
    float sc = r[5 + c];
    if (sc > m) { m = sc; mi = c; }
  }
  float conf = m * obj;
  int v = (obj > conf_thres) && (conf > conf_thres);
  float cx = r[0], cy = r[1], w = r[2], h = r[3];
  float x1 = cx - w * 0.5f, y1 = cy - h * 0.5f;
  float x2 = cx + w * 0.5f, y2 = cy + h * 0.5f;
  size_t o = ((size_t)b * K + t) * 4;
  boxes[o + 0] = x1; boxes[o + 1] = y1; boxes[o + 2] = x2; boxes[o + 3] = y2;
  cls[(size_t)b * K + t] = (float)mi;
  valid[(size_t)b * K + t] = v;
  float off = (float)mi * MAX_WH;  // class-offset trick for NMS
  obox[o + 0] = x1 + off; obox[o + 1] = y1 + off;
  obox[o + 2] = x2 + off; obox[o + 3] = y2 + off;
  oarea[(size_t)b * K + t] = (x2 - x1) * (y2 - y1);
}

// ---------------------------------------------------------------------------
// Stage 4a: suppression-bit matrix. Thread (b, r, c) packs bits for columns
// j = 32c..32c+31 of row r. Division-free: iou>t <=> inter*(1+t) > t*(ai+aj).
// ---------------------------------------------------------------------------
__global__ void nms_pairs_kernel(const float* __restrict__ obox,
                                 const float* __restrict__ oarea,
                                 int K, unsigned* __restrict__ smat,
                                 int total) {
  int id = blockIdx.x * blockDim.x + threadIdx.x;
  if (id >= total) return;
  int c = id & 31;
  int rest = id >> 5;
  int r = rest % K;
  int b = rest / K;
  unsigned bits = 0u;
  int chunks = K >> 5;
  if (c < chunks) {
    size_t ro = ((size_t)b * K + r) * 4;
    float x1 = obox[ro + 0], y1 = obox[ro + 1];
    float x2 = obox[ro + 2], y2 = obox[ro + 3];
    float ai = oarea[(size_t)b * K + r];
    int jbase = c << 5;
#pragma unroll 8
    for (int jj = 0; jj < 32; ++jj) {
      int j = jbase + jj;
      size_t jo = ((size_t)b * K + j) * 4;
      float u1 = fmaxf(x1, obox[jo + 0]), w1 = fmaxf(y1, obox[jo + 1]);
      float u2 = fminf(x2, obox[jo + 2]), w2 = fminf(y2, obox[jo + 3]);
      float iw = fmaxf(u2 - u1, 0.0f), ih = fmaxf(w2 - w1, 0.0f);
      float inter = iw * ih;
      bool sup = (j > r) &&
                 (inter * (1.0f + IOU_THR) >
                  IOU_THR * (ai + oarea[(size_t)b * K + j]));
      bits |= (sup ? 1u : 0u) << jj;
    }
  }
  smat[((size_t)b * K + r) * 32 + c] = bits;
}

// ---------------------------------------------------------------------------
// Stage 4b: greedy sweep, ONE wave per image. Lane l owns keep-bits for boxes
// 32l..32l+31 in a single VGPR. 8-deep software-pipelined row prefetch ring
// keeps 8 loads in flight to hide L2 latency of the serial walk.
// ---------------------------------------------------------------------------
__global__ void nms_sweep_kernel(const int* __restrict__ valid,
                                 const unsigned* __restrict__ smat,
                                 int K, int* __restrict__ keep_out) {
  int b = blockIdx.x;
  int lane = threadIdx.x;  // 32 threads
  int chunks = K >> 5;
  unsigned keepm = 0u;
  if (lane < chunks) {
    const int* vv = valid + (size_t)b * K + (lane << 5);
#pragma unroll 8
    for (int j = 0; j < 32; ++j) keepm |= (vv[j] ? 1u : 0u) << j;
  }
  const unsigned* base = smat + ((size_t)b * K) * 32;
  unsigned buf[8];
#pragma unroll
  for (int jj = 0; jj < 8; ++jj) buf[jj] = base[jj * 32 + lane];
  for (int blk = 0; blk < (K >> 3); ++blk) {
#pragma unroll
    for (int jj = 0; jj < 8; ++jj) {
      int i = (blk << 3) + jj;
      unsigned cur = buf[jj];
      int pre = i + 8;
      buf[jj] = (pre < K) ? base[pre * 32 + lane] : 0u;  // prefetch ahead
      int owner = i >> 5, bit = i & 31;
      int ki = (__shfl((int)keepm, owner, 32) >> bit) & 1;
      if (ki) keepm &= ~cur;  // uniform branch (broadcast value)
    }
  }
  if (lane < chunks) {
    int* ko = keep_out + (size_t)b * K + (lane << 5);
#pragma unroll 8
    for (int j = 0; j < 32; ++j) ko[j] = (int)((keepm >> j) & 1u);
  }
}

// ---------------------------------------------------------------------------
// Stage 5: cross-match patch(1024) x clean(256), normalized boxes.
// Denominator area_p[m]+area_c[n] is a rank-2 outer-sum matmul computed per
// 16x16 tile with v_wmma_f32_16x16x4_f32:
//   A(16x4): col0 = area_p tile, col1 = ones   (lanes 0-15: {a,1}, 16-31: 0)
//   B(4x16): row0 = ones, row1 = area_c tile   (lanes 0-15: {1,a}, 16-31: 0)
// D layout: lane l, reg r -> (M = r + 8*(l>>4), N = l&15), matched below.
// Branchless: keep+class fused into one "effective class" LDS word
// (kept ? cls : -1e9) so the match mask is one load + cmp + cndmask; WMMA
// operand builds use unconditional clamped loads + selects (no exec churn).
// ---------------------------------------------------------------------------
__global__ void match_kernel(const float* __restrict__ bp,
                             const float* __restrict__ cp,
                             const int* __restrict__ kp,
                             const float* __restrict__ bc,
                             const float* __restrict__ cc,
                             const int* __restrict__ kc,
                             float* __restrict__ sums,
                             float* __restrict__ counts) {
  __shared__ float sbp[KP * 4];
  __shared__ float sap[KP];
  __shared__ float seff[KP];  // kept ? cls : -1e9  (never equals a real cls)
  __shared__ float sbc[KC * 4];
  __shared__ float sac[KC];
  __shared__ float scc[KC];
  __shared__ int   skc[KC];
  __shared__ int   stm[KC];   // per-column max as float bits (iou >= 0)
  __shared__ float rs[8], rc2[8];

  int b = blockIdx.x, t = threadIdx.x;  // 256 threads = 8 waves
  for (int k = t; k < KP; k += 256) {
    size_t o = ((size_t)b * KP + k) * 4;
    float x1 = bp[o + 0] * INV_GN, y1 = bp[o + 1] * INV_GN;
    float x2 = bp[o + 2] * INV_GN, y2 = bp[o + 3] * INV_GN;
    sbp[k * 4 + 0] = x1; sbp[k * 4 + 1] = y1;
    sbp[k * 4 + 2] = x2; sbp[k * 4 + 3] = y2;
    sap[k] = (x2 - x1) * (y2 - y1);
    seff[k] = kp[(size_t)b * KP + k] ? cp[(size_t)b * KP + k] : -1e9f;
  }
  {
    size_t o = ((size_t)b * KC + t) * 4;
    float x1 = bc[o + 0] * INV_GN, y1 = bc[o + 1] * INV_GN;
    float x2 = bc[o + 2] * INV_GN, y2 = bc[o + 3] * INV_GN;
    sbc[t * 4 + 0] = x1; sbc[t * 4 + 1] = y1;
    sbc[t * 4 + 2] = x2; sbc[t * 4 + 3] = y2;
    sac[t] = (x2 - x1) * (y2 - y1);
    scc[t] = cc[(size_t)b * KC + t];
    skc[t] = kc[(size_t)b * KC + t];
    stm[t] = 0;
  }
  __syncthreads();

  int lane = t & 31, wave = t >> 5;
  int l15 = lane & 15;
  float laneSel = (lane < 16) ? 1.0f : 0.0f;  // cndmask, applied to operands
  for (int ct = wave; ct < 16; ct += 8) {  // 2 clean tiles per wave
    int n = ct * 16 + l15;
    float cx1 = sbc[n * 4 + 0], cy1 = sbc[n * 4 + 1];
    float cx2 = sbc[n * 4 + 2], cy2 = sbc[n * 4 + 3];
    float ccn = scc[n];
    v2f B2;
    B2.x = laneSel;                      // row0 = ones
    B2.y = laneSel * sac[ct * 16 + l15]; // row1 = area_c (0 for lanes 16-31)
    float colmax = 0.0f;
    for (int pt = 0; pt < 64; ++pt) {
      v2f A2;
      A2.x = laneSel * sap[pt * 16 + l15];  // col0 = area_p
      A2.y = laneSel;                       // col1 = ones
      v8f C;
#pragma unroll
      for (int q = 0; q < 8; ++q) C[q] = 0.0f;
      v8f D = __builtin_amdgcn_wmma_f32_16x16x4_f32(
          false, A2, false, B2, (short)0, C, false, false);
      int mbase = pt * 16 + ((lane >> 4) << 3);
#pragma unroll
      for (int r0 = 0; r0 < 8; ++r0) {
        int m = mbase + r0;
        float px1 = sbp[m * 4 + 0], py1 = sbp[m * 4 + 1];
        float px2 = sbp[m * 4 + 2], py2 = sbp[m * 4 + 3];
        float tlx = fmaxf(px1, cx1), tly = fmaxf(py1, cy1);
        float brx = fminf(px2, cx2), bry = fminf(py2, cy2);
        float iw = fmaxf(brx - tlx, 0.0f), ih = fmaxf(bry - tly, 0.0f);
        float inter = iw * ih;
        float iou = inter * __builtin_amdgcn_rcpf(D[r0] - inter);
        float val = (seff[m] == ccn) ? iou : 0.0f;  // one load + cmp + cndmask
        colmax = fmaxf(colmax, val);
      }
    }
    atomicMax(&stm[n], __float_as_int(colmax));
  }
  __syncthreads();

  float contrib = skc[t] ? __int_as_float(stm[t]) : 0.0f;
  float cnt = skc[t] ? 1.0f : 0.0f;
#pragma unroll
  for (int offn = 16; offn > 0; offn >>= 1) {
    contrib += __shfl_down(contrib, offn, 32);
    cnt += __shfl_down(cnt, offn, 32);
  }
  if (lane == 0) { rs[wave] = contrib; rc2[wave] = cnt; }
  __syncthreads();
  if (t == 0) {
    float s = 0.0f, c2 = 0.0f;
#pragma unroll
    for (int wv = 0; wv < 8; ++wv) { s += rs[wv]; c2 += rc2[wv]; }
    sums[b] = s;
    counts[b] = c2;
  }
}

__global__ void finalize_kernel(const float* __restrict__ sums,
                                const float* __restrict__ counts,
                                float* __restrict__ out) {
  if (threadIdx.x == 0) {
    float s = 0.0f, c = 0.0f;
#pragma unroll
    for (int b = 0; b < NB; ++b) { s += sums[b]; c += counts[b]; }
    out[0] = (c > 0.0f) ? (1.0f - s / fmaxf(c, 1.0f)) : 1.0f;
  }
}

// ---------------------------------------------------------------------------
extern "C" void kernel_launch(void* const* d_in, const int* in_sizes, int n_in,
                              void* d_out, int out_size, void* d_ws,
                              size_t ws_size, hipStream_t stream) {
  const float* clean = (const float*)d_in[0];
  const float* patch = (const float*)d_in[1];
  float* out = (float*)d_out;

  // workspace carve (all 4-byte elements)
  float* msC  = (float*)d_ws;                     // NB*NN
  float* msP  = msC + (size_t)NB * NN;            // NB*NN
  int*   idxC = (int*)(msP + (size_t)NB * NN);    // NB*KP
  int*   idxP = idxC + NB * KP;                   // NB*KP
  float* bcB  = (float*)(idxP + NB * KP);         // NB*KC*4
  float* ccB  = bcB + NB * KC * 4;                // NB*KC
  int*   vcB  = (int*)(ccB + NB * KC);            // NB*KC
  float* obC  = (float*)(vcB + NB * KC);          // NB*KC*4
  float* oaC  = obC + NB * KC * 4;                // NB*KC
  float* bpB  = oaC + NB * KC;                    // NB*KP*4
  float* cpB  = bpB + NB * KP * 4;                // NB*KP
  int*   vpB  = (int*)(cpB + NB * KP);            // NB*KP
  float* obP  = (float*)(vpB + NB * KP);          // NB*KP*4
  float* oaP  = obP + NB * KP * 4;                // NB*KP
  int*   kcB  = (int*)(oaP + NB * KP);            // NB*KC
  int*   kpB  = kcB + NB * KC;                    // NB*KP
  unsigned* smC = (unsigned*)(kpB + NB * KP);     // NB*KC*32
  unsigned* smP = smC + (size_t)NB * KC * 32;     // NB*KP*32
  float* sums = (float*)(smP + (size_t)NB * KP * 32);  // NB
  float* cnts = sums + NB;                        // NB

  const int total = NB * NN;
  const int thr = 256;
  score_kernel<<<(total + thr - 1) / thr, thr, 0, stream>>>(clean, msC, 0.25f, total);
  score_kernel<<<(total + thr - 1) / thr, thr, 0, stream>>>(patch, msP, 0.001f, total);

  topk_kernel<<<NB, 1024, 0, stream>>>(msC, idxC);
  topk_kernel<<<NB, 1024, 0, stream>>>(msP, idxP);

  gather_kernel<<<NB, KC, 0, stream>>>(clean, idxC, 0.25f, bcB, ccB, vcB, obC, oaC);
  gather_kernel<<<NB, KP, 0, stream>>>(patch, idxP, 0.001f, bpB, cpB, vpB, obP, oaP);

  const int totC = NB * KC * 32;
  const int totP = NB * KP * 32;
  nms_pairs_kernel<<<(totC + thr - 1) / thr, thr, 0, stream>>>(obC, oaC, KC, smC, totC);
  nms_pairs_kernel<<<(totP + thr - 1) / thr, thr, 0, stream>>>(obP, oaP, KP, smP, totP);
  nms_sweep_kernel<<<NB, 32, 0, stream>>>(vcB, smC, KC, kcB);
  nms_sweep_kernel<<<NB, 32, 0, stream>>>(vpB, smP, KP, kpB);

  match_kernel<<<NB, 256, 0, stream>>>(bpB, cpB, kpB, bcB, ccB, kcB, sums, cnts);
  finalize_kernel<<<1, 32, 0, stream>>>(sums, cnts, out);
}